// YoloLoss_11982958756709
// MI455X (gfx1250) — compile-verified
//
#include <hip/hip_runtime.h>

// ---------------- problem constants (from reference) ----------------
#define SDIM        14
#define NCLS        20
#define CELL_STRIDE 30          // B*5 + NUM_CLS = 2*5 + 20
#define L_COORD     5.0f
#define L_NOOBJ     0.5f

typedef __attribute__((ext_vector_type(2))) float v2f;
typedef __attribute__((ext_vector_type(8))) float v8f;

// =====================================================================
// Kernel 1: streaming per-cell loss, block-level partial sums.
// Accumulates 4 raw components per thread:
//   acc0 = sum of masked (pred_cls - target_cls)^2          -> cls_loss
//   acc1 = sum of unmasked (conf0^2 + conf1^2)              -> no_obj = 0.5*acc1
//   acc2 = sum of masked (best_conf - best_iou)^2           -> contain_loss
//   acc3 = sum of masked (dxy + dwh)                        -> reg = 5*acc3
// Deterministic: fixed grid, fixed per-thread visit order, tree reduce.
// =====================================================================
__global__ void __launch_bounds__(256)
yolo_loss_main(const float* __restrict__ pred,
               const float* __restrict__ tbox,
               const float* __restrict__ tcls,
               const unsigned char* __restrict__ mask,
               float* __restrict__ partials,
               int nCells)
{
    const float invS = 1.0f / (float)SDIM;
    float acc0 = 0.0f, acc1 = 0.0f, acc2 = 0.0f, acc3 = 0.0f;

    for (int cell = blockIdx.x * blockDim.x + threadIdx.x;
         cell < nCells;
         cell += gridDim.x * blockDim.x)
    {
        // ---- loads: widest legal vectors (120B stride -> 8B aligned) ----
        const float2* p2 = (const float2*)(pred + (size_t)cell * CELL_STRIDE);
        float2 pv[15];
        #pragma unroll
        for (int j = 0; j < 15; ++j) pv[j] = p2[j];          // global_load_b64 x15

        float4 tb = ((const float4*)tbox)[cell];             // global_load_b128

        const float4* tc4 = (const float4*)(tcls + (size_t)cell * NCLS);
        float4 tcv[5];
        #pragma unroll
        for (int j = 0; j < 5; ++j) tcv[j] = tc4[j];         // global_load_b128 x5

        const float m  = mask[cell] ? 1.0f : 0.0f;
        const float nm = 1.0f - m;

        // ---- unpack pred boxes: [x0 y0 w0 h0 c0 x1 y1 w1 h1 c1] ----
        const float bx0 = pv[0].x, by0 = pv[0].y, bw0 = pv[1].x, bh0 = pv[1].y, bc0 = pv[2].x;
        const float bx1 = pv[2].y, by1 = pv[3].x, bw1 = pv[3].y, bh1 = pv[4].x, bc1 = pv[4].y;

        // ---- class loss ----
        float pc[NCLS], tc[NCLS];
        #pragma unroll
        for (int j = 0; j < 10; ++j) { pc[2*j] = pv[5 + j].x; pc[2*j + 1] = pv[5 + j].y; }
        #pragma unroll
        for (int j = 0; j < 5; ++j) {
            tc[4*j + 0] = tcv[j].x; tc[4*j + 1] = tcv[j].y;
            tc[4*j + 2] = tcv[j].z; tc[4*j + 3] = tcv[j].w;
        }
        float cls = 0.0f;
        #pragma unroll
        for (int k = 0; k < NCLS; ++k) { const float d = pc[k] - tc[k]; cls = fmaf(d, d, cls); }
        acc0 = fmaf(m, cls, acc0);

        // ---- no-object confidence loss (raw, *0.5 at the end) ----
        acc1 = fmaf(nm, bc0 * bc0 + bc1 * bc1, acc1);

        // ---- target box to xyxy ----
        const float tcx = tb.x * invS, tcy = tb.y * invS;
        const float thw = 0.5f * tb.z, thh = 0.5f * tb.w;
        const float t0 = tcx - thw, t1 = tcy - thh, t2 = tcx + thw, t3 = tcy + thh;
        const float ta = (t2 - t0) * (t3 - t1);

        // ---- IoU box 0 ----
        float cx = bx0 * invS, cy = by0 * invS, hw = 0.5f * bw0, hh = 0.5f * bh0;
        float q0 = cx - hw, q1 = cy - hh, q2 = cx + hw, q3 = cy + hh;
        float w  = fmaxf(fminf(q2, t2) - fmaxf(q0, t0), 0.0f);
        float h  = fmaxf(fminf(q3, t3) - fmaxf(q1, t1), 0.0f);
        float inter = w * h;
        float pa = (q2 - q0) * (q3 - q1);
        const float iou0 = inter / (pa + ta - inter);

        // ---- IoU box 1 ----
        cx = bx1 * invS; cy = by1 * invS; hw = 0.5f * bw1; hh = 0.5f * bh1;
        q0 = cx - hw; q1 = cy - hh; q2 = cx + hw; q3 = cy + hh;
        w  = fmaxf(fminf(q2, t2) - fmaxf(q0, t0), 0.0f);
        h  = fmaxf(fminf(q3, t3) - fmaxf(q1, t1), 0.0f);
        inter = w * h;
        pa = (q2 - q0) * (q3 - q1);
        const float iou1 = inter / (pa + ta - inter);

        // ---- best box (argmax; first index wins ties like jnp.argmax) ----
        const bool  sel  = iou1 > iou0;
        const float bbx  = sel ? bx1 : bx0;
        const float bby  = sel ? by1 : by0;
        const float bbw  = sel ? bw1 : bw0;
        const float bbh  = sel ? bh1 : bh0;
        const float bbc  = sel ? bc1 : bc0;
        const float biou = sel ? iou1 : iou0;

        // ---- regression (raw, *5 at the end) + containment ----
        const float dx = bbx - tb.x, dy = bby - tb.y;
        const float dw = sqrtf(bbw) - sqrtf(tb.z);
        const float dh = sqrtf(bbh) - sqrtf(tb.w);
        acc3 = fmaf(m, fmaf(dx, dx, dy * dy) + fmaf(dw, dw, dh * dh), acc3);
        const float dc = bbc - biou;
        acc2 = fmaf(m, dc * dc, acc2);
    }

    // ---- wave32 tree reduction ----
    #pragma unroll
    for (int off = 16; off >= 1; off >>= 1) {
        acc0 += __shfl_xor(acc0, off, 32);
        acc1 += __shfl_xor(acc1, off, 32);
        acc2 += __shfl_xor(acc2, off, 32);
        acc3 += __shfl_xor(acc3, off, 32);
    }

    __shared__ float red[8][4];
    const int lane = threadIdx.x & 31;
    const int wave = threadIdx.x >> 5;
    if (lane == 0) {
        red[wave][0] = acc0; red[wave][1] = acc1;
        red[wave][2] = acc2; red[wave][3] = acc3;
    }
    __syncthreads();

    if (wave == 0) {
        float r0 = (lane < 8) ? red[lane][0] : 0.0f;
        float r1 = (lane < 8) ? red[lane][1] : 0.0f;
        float r2 = (lane < 8) ? red[lane][2] : 0.0f;
        float r3 = (lane < 8) ? red[lane][3] : 0.0f;
        #pragma unroll
        for (int off = 4; off >= 1; off >>= 1) {
            r0 += __shfl_xor(r0, off, 32);
            r1 += __shfl_xor(r1, off, 32);
            r2 += __shfl_xor(r2, off, 32);
            r3 += __shfl_xor(r3, off, 32);
        }
        if (lane == 0) {
            float4* pout = (float4*)(partials + (size_t)blockIdx.x * 4);
            *pout = make_float4(r0, r1, r2, r3);
        }
    }
}

// =====================================================================
// Kernel 2: single-wave finisher. Deterministic strided accumulation of
// per-block partials, then the final cross-lane sum is done on the
// matrix unit: V_WMMA_F32_16X16X4_F32 with B = ones computes
// D[c,n] = sum_k A[c,k]; we place the 4 group-sums of component c in
// A-matrix row c (per the ISA 16x4 A layout: lane L holds K=0..1,
// lane L+16 holds K=2..3). Lane 0 then reads D[c,0] from VGPRs 0..3.
// EXEC is all ones at the WMMA (whole wave executes it).
// =====================================================================
__global__ void __launch_bounds__(32)
yolo_loss_final(const float* __restrict__ partials,
                float* __restrict__ out,
                int nBlocks)
{
    const int lane = threadIdx.x;   // exactly 32 threads
    float acc[4] = {0.0f, 0.0f, 0.0f, 0.0f};

    for (int i = lane; i < nBlocks; i += 32) {
        const float4 p = ((const float4*)partials)[i];
        acc[0] += p.x; acc[1] += p.y; acc[2] += p.z; acc[3] += p.w;
    }

    // reduce within groups of 8 lanes -> 4 group sums per component
    #pragma unroll
    for (int off = 1; off <= 4; off <<= 1) {
        acc[0] += __shfl_xor(acc[0], off, 32);
        acc[1] += __shfl_xor(acc[1], off, 32);
        acc[2] += __shfl_xor(acc[2], off, 32);
        acc[3] += __shfl_xor(acc[3], off, 32);
    }

    // G[g][c]: group-g sum of component c
    __shared__ float G[16];
    if ((lane & 7) == 0) {
        const int g = lane >> 3;
        G[g * 4 + 0] = acc[0]; G[g * 4 + 1] = acc[1];
        G[g * 4 + 2] = acc[2]; G[g * 4 + 3] = acc[3];
    }
    __syncthreads();

    // Build A (16x4): A[c, k] = G[k][c] for c<4, zero elsewhere.
    // ISA layout: lane L(0..15) -> {A[L,0], A[L,1]}, lane L+16 -> {A[L,2], A[L,3]}
    const int  L  = lane & 15;
    const int  kb = (lane >= 16) ? 2 : 0;
    float ax = 0.0f, ay = 0.0f;
    if (L < 4) {
        ax = G[(kb + 0) * 4 + L];
        ay = G[(kb + 1) * 4 + L];
    }
    v2f A;  A[0] = ax;   A[1] = ay;
    v2f Bm; Bm[0] = 1.0f; Bm[1] = 1.0f;       // all-ones B => D = row sums of A
    v8f C = {};

    v8f D = __builtin_amdgcn_wmma_f32_16x16x4_f32(
        /*neg_a=*/false, A, /*neg_b=*/false, Bm,
        /*c_mod=*/(short)0, C, /*reuse_a=*/false, /*reuse_b=*/false);

    if (lane == 0) {
        const float cls    = D[0];             // D[0,0]
        const float conf2  = D[1];             // D[1,0]
        const float con    = D[2];             // D[2,0]
        const float regraw = D[3];             // D[3,0]
        const float reg    = L_COORD * regraw;
        const float noobj  = L_NOOBJ * conf2;
        out[0] = cls + noobj + con + reg;      // total
        out[1] = reg;                          // reg_loss
        out[2] = con;                          // contain_loss
        out[3] = noobj;                        // no_obj_loss
        out[4] = cls;                          // cls_loss
    }
}

// =====================================================================
// Host launcher
// =====================================================================
extern "C" void kernel_launch(void* const* d_in, const int* in_sizes, int n_in,
                              void* d_out, int out_size, void* d_ws, size_t ws_size,
                              hipStream_t stream)
{
    const float*         pred = (const float*)d_in[0];  // (N,S,S,30) f32
    const float*         tbox = (const float*)d_in[1];  // (N,S,S,4)  f32
    const float*         tcls = (const float*)d_in[2];  // (N,S,S,20) f32
    const unsigned char* msk  = (const unsigned char*)d_in[3]; // (N,S,S) bool
    float*               out  = (float*)d_out;          // 5 scalars
    float*               ws   = (float*)d_ws;           // nBlocks*4 partials

    const int nCells  = in_sizes[1] / 4;                // target_boxes: 4 floats/cell
    int nBlocks = (nCells + 255) / 256;
    if (nBlocks > 1024) nBlocks = 1024;                 // grid-stride cap; 16 KB ws
    if (nBlocks < 1) nBlocks = 1;

    yolo_loss_main<<<nBlocks, 256, 0, stream>>>(pred, tbox, tcls, msk, ws, nCells);
    yolo_loss_final<<<1, 32, 0, stream>>>(ws, out, nBlocks);
}